// MultiBitCascadedExtractor_62380105007352
// MI455X (gfx1250) — compile-verified
//
#include <hip/hip_runtime.h>
#include <cstdint>

// MultiBitCascadedExtractor for MI455X (gfx1250).
// Closed form of the reference cascade: m = max(int(value)-1, 0);
//   all_bits[b,i]  = (m>>i)&1  (f32)
//   selected[L][b] = (m>>(14-2L))&3 (i32)
// Memory-bound streaming: ~400 MB traffic -> ~17us floor @ 23.3 TB/s.
// CDNA5 path: TDM tensor_load_to_lds per wave + s_wait_tensorcnt, NT b128 stores.

typedef __attribute__((ext_vector_type(4))) unsigned v4u;
typedef __attribute__((ext_vector_type(8))) unsigned v8u;
typedef __attribute__((ext_vector_type(4))) float    v4f;
typedef __attribute__((ext_vector_type(4))) int      v4i;

#define TOTAL_BITS 16
#define NUM_LAYERS 8
#define BLOCK 256
#define EPT 4                         // elements per thread
#define EPB (BLOCK * EPT)             // 1024 elements per block
#define WAVE_ELEMS (32 * EPT)         // 128 elements per wave tile
#define WAVE_BYTES (WAVE_ELEMS * 4)   // 512 B per wave TDM descriptor

__device__ __forceinline__ unsigned mapped_value(float fv) {
  int iv = (int)fv;                       // inputs are integral-valued f32
  return iv > 0 ? (unsigned)(iv - 1) : 0u;
}

__global__ __launch_bounds__(BLOCK)
void MultiBitCascadedExtractor_62380105007352_kernel(const float* __restrict__ in,
                                                     float* __restrict__ out_bits,
                                                     int* __restrict__ out_sel,
                                                     int n) {
  __shared__ __align__(16) float tile[EPB];
  const int tid = (int)threadIdx.x;
  const long long blockStart = (long long)blockIdx.x * EPB;
  const long long e0 = blockStart + (long long)tid * EPT;

  if (blockStart + EPB <= (long long)n) {
    // ---------- full tile: TDM DMA global -> LDS, one descriptor per wave ----------
    unsigned wave = (unsigned)__builtin_amdgcn_readfirstlane((int)((unsigned)tid >> 5));
    unsigned long long gaddr =
        (unsigned long long)(const void*)(in + blockStart) +
        (unsigned long long)wave * WAVE_BYTES;
    // low 32 bits of a generic LDS pointer == workgroup-relative LDS byte offset
    unsigned lds_addr = (unsigned)(unsigned long long)&tile[0] + wave * WAVE_BYTES;

    // D# group 0 (ISA 8.3): count=1 | lds_addr | global_addr[56:0] | type=2
    v4u g0 = { 1u,
               lds_addr,
               (unsigned)(gaddr & 0xFFFFFFFFull),
               (unsigned)(gaddr >> 32) | 0x80000000u };
    // D# group 1 (ISA 8.4): data_size=2 (4B), tensor_dim0=128, tensor_dim1=1,
    // tile_dim0=128 (1-D tile), tensor_dim0_stride=128, no pad/multicast/iterate.
    v8u g1 = { 0x00020000u,
               (unsigned)WAVE_ELEMS << 16,   // tensor_dim0[15:0] @ bits 63:48
               1u << 16,                     // tensor_dim1[15:0] @ bits 95:80
               (unsigned)WAVE_ELEMS << 16,   // tile_dim0 @ bits 127:112
               0u,                           // tile_dim1/tile_dim2 unused
               (unsigned)WAVE_ELEMS,         // tensor_dim0_stride[31:0]
               0u, 0u };
    v4u g2 = { 0u, 0u, 0u, 0u };            // <=2D tensor: groups 2/3 inert
    v4u g3 = { 0u, 0u, 0u, 0u };

    asm volatile("tensor_load_to_lds %0, %1, %2, %3"
                 :: "s"(g0), "s"(g1), "s"(g2), "s"(g3)
                 : "memory");
    asm volatile("s_wait_tensorcnt 0x0" ::: "memory");

    // Each thread reads only from its own wave's 512B sub-tile -> no barrier.
    v4f v = ((const v4f*)tile)[tid];
    unsigned m[EPT] = { mapped_value(v.x), mapped_value(v.y),
                        mapped_value(v.z), mapped_value(v.w) };

    // all_bits: [B,16] f32. Thread writes 256B contiguous via 16 NT b128 stores.
    float* bp = out_bits + e0 * TOTAL_BITS;
#pragma unroll
    for (int k = 0; k < EPT; ++k) {
#pragma unroll
      for (int c = 0; c < 4; ++c) {
        v4f b = { (float)((m[k] >> (4 * c + 0)) & 1u),
                  (float)((m[k] >> (4 * c + 1)) & 1u),
                  (float)((m[k] >> (4 * c + 2)) & 1u),
                  (float)((m[k] >> (4 * c + 3)) & 1u) };
        __builtin_nontemporal_store(b, (v4f*)(bp + k * TOTAL_BITS + c * 4));
      }
    }
    // selected: [8,B] i32. Per layer one lane-contiguous NT v4i store.
#pragma unroll
    for (int L = 0; L < NUM_LAYERS; ++L) {
      const int sh = 14 - 2 * L;
      v4i s = { (int)((m[0] >> sh) & 3u), (int)((m[1] >> sh) & 3u),
                (int)((m[2] >> sh) & 3u), (int)((m[3] >> sh) & 3u) };
      __builtin_nontemporal_store(s, (v4i*)(out_sel + (long long)L * n + e0));
    }
  } else {
    // ---------- tail (n not a multiple of EPB): scalar fallback ----------
    for (int k = 0; k < EPT; ++k) {
      long long e = e0 + k;
      if (e >= (long long)n) break;
      unsigned mm = mapped_value(in[e]);
      for (int i = 0; i < TOTAL_BITS; ++i)
        out_bits[e * TOTAL_BITS + i] = (float)((mm >> i) & 1u);
      for (int L = 0; L < NUM_LAYERS; ++L)
        out_sel[(long long)L * n + e] = (int)((mm >> (14 - 2 * L)) & 3u);
    }
  }
}

extern "C" void kernel_launch(void* const* d_in, const int* in_sizes, int n_in,
                              void* d_out, int out_size, void* d_ws, size_t ws_size,
                              hipStream_t stream) {
  const float* value = (const float*)d_in[0];
  const int n = in_sizes[0];
  float* out_bits = (float*)d_out;                                  // [B,16] f32
  int*   out_sel  = (int*)((float*)d_out + (size_t)n * TOTAL_BITS); // [8,B] i32
  const int grid = (n + EPB - 1) / EPB;
  MultiBitCascadedExtractor_62380105007352_kernel<<<grid, BLOCK, 0, stream>>>(
      value, out_bits, out_sel, n);
  (void)n_in; (void)out_size; (void)d_ws; (void)ws_size;
}